// NeuralODECortex_63960652972782
// MI455X (gfx1250) — compile-verified
//
#include <hip/hip_runtime.h>
#include <math.h>

// CDNA5 / gfx1250: wave32, WMMA fp32 path (V_WMMA_F32_16X16X4_F32).
typedef __attribute__((ext_vector_type(2))) float v2f;
typedef __attribute__((ext_vector_type(8))) float v8f;

#define WAVES 8          // waves (tiles) per block
#define TILE  16         // batch rows per wave  (M of WMMA)
#define LDA   68         // padded LDS row stride (floats): conflict-free scatter+gather

__device__ __forceinline__ v8f wmma4(v2f a, v2f b, v8f c) {
    // D = A(16x4) * B(4x16) + C(16x16), all f32
    return __builtin_amdgcn_wmma_f32_16x16x4_f32(
        /*neg_a=*/false, a, /*neg_b=*/false, b,
        /*c_mod=*/(short)0, c, /*reuse_a=*/false, /*reuse_b=*/false);
}

__device__ __forceinline__ float fast_tanh(float x) {
#if __has_builtin(__builtin_amdgcn_tanhf)
    return __builtin_amdgcn_tanhf(x);          // v_tanh_f32 (trans op)
#else
    return tanhf(x);
#endif
}

__global__ __launch_bounds__(WAVES * 32)
void node_rk45_kernel(const float* __restrict__ pad0,
                      const float* __restrict__ sens,
                      const float* __restrict__ W1, const float* __restrict__ b1,
                      const float* __restrict__ W2, const float* __restrict__ b2,
                      const float* __restrict__ W3, const float* __restrict__ b3,
                      const float* __restrict__ scale_p,
                      float* __restrict__ out, int ntiles)
{
    // Swizzled W2 B-fragments: [kchunk 16][ntile 4][lane 32] -> float2 (16 KB)
    __shared__ float2 w2s[16 * 4 * 32];
    // Per-wave activation staging, row-major [16][64] padded to stride 68 (~4.25 KB each)
    __shared__ float astage[WAVES][TILE * LDA];
    // Per-wave k staging: [16 rows][4 comps] (col 3 = zero pad)
    __shared__ float kstage[WAVES][TILE * 4];

    const int tid  = threadIdx.x;
    const int wave = tid >> 5;
    const int lane = tid & 31;
    const int half = lane >> 4;     // 0: lanes 0-15, 1: lanes 16-31
    const int l    = lane & 15;

    // ---- one-time: swizzle W2 (64x64 row-major) into B-fragment layout ----
    for (int idx = tid; idx < 16 * 4 * 32; idx += WAVES * 32) {
        int c  = idx >> 7;
        int n  = (idx >> 5) & 3;
        int ln = idx & 31;
        int hh = ln >> 4, ll = ln & 15;
        int r0 = 4 * c + 2 * hh;          // B rows (K) held by this lane
        int col = n * 16 + ll;            // B col (N)
        w2s[idx] = make_float2(W2[r0 * 64 + col], W2[(r0 + 1) * 64 + col]);
    }
    __syncthreads();

    const int tile = blockIdx.x * WAVES + wave;
    if (tile >= ntiles) return;                 // wave-uniform
    const int  base  = tile * TILE;
    const float scale = scale_p[0];
    const float hstep = 0.1f;                   // TDELTA / NSTEPS

    float* mystage = astage[wave];
    float* myk     = kstage[wave];

    // ---- constant per-lane fragments ----
    // Layer-1 B fragments: rows {W1[0], W1[1], W1[2], W1[64]} (y0,y1,y2,t features)
    v2f w1f[4];
#pragma unroll
    for (int n = 0; n < 4; ++n) {
        int col = n * 16 + l;
        w1f[n].x = half ? W1[2 * 64 + col] : W1[0 * 64 + col];
        w1f[n].y = half ? W1[64 * 64 + col] : W1[1 * 64 + col];
    }
    // Layer-3 B fragments: W3 (64x3) zero-padded to 64x16
    v2f w3f[16];
#pragma unroll
    for (int c = 0; c < 16; ++c) {
        int r0 = 4 * c + 2 * half;
        w3f[c].x = (l < 3) ? W3[r0 * 3 + l]       : 0.f;
        w3f[c].y = (l < 3) ? W3[(r0 + 1) * 3 + l] : 0.f;
    }
    float b1v[4], b2v[4];
#pragma unroll
    for (int n = 0; n < 4; ++n) { b1v[n] = b1[n * 16 + l]; b2v[n] = b2[n * 16 + l]; }
    const float b3v = (l < 3) ? b3[l] : 0.f;

    // ---- stage sensory tile (padded K 61->64) and precompute S_c = b1 + sens@W1[3:64] ----
    for (int idx = lane; idx < TILE * 64; idx += 32) {
        int r = idx >> 6, cc = idx & 63;
        mystage[r * LDA + cc] = (cc < 61) ? sens[(base + r) * 61 + cc] : 0.f;
    }
    v8f Sc[4];
#pragma unroll
    for (int n = 0; n < 4; ++n) {
        v8f z;
#pragma unroll
        for (int v = 0; v < 8; ++v) z[v] = b1v[n];
        Sc[n] = z;
    }
#pragma unroll
    for (int c = 0; c < 16; ++c) {
        v2f a = *(const v2f*)&mystage[l * LDA + 4 * c + 2 * half];
        int r0 = 4 * c + 2 * half;        // sensory feature row (0..63, valid 0..60)
#pragma unroll
        for (int n = 0; n < 4; ++n) {
            int col = n * 16 + l;
            v2f b;
            b.x = (r0     <= 60) ? W1[(3 + r0) * 64 + col] : 0.f;
            b.y = (r0 + 1 <= 60) ? W1[(4 + r0) * 64 + col] : 0.f;
            Sc[n] = wmma4(a, b, Sc[n]);
        }
    }

    // ---- initial state: lanes 0-15 hold (y0,y1) of row l; lanes 16-31 hold (y2, -) ----
    float2 y;
    {
        const float* p = pad0 + (base + l) * 3;
        y.x = half ? p[2] : p[0];
        y.y = half ? 0.f  : p[1];
    }

    // ---- one MLP eval: ya -> k (per-lane float2, same lane convention) ----
    auto feval = [&](float tt, float2 ya) -> float2 {
        // Layer 1: rank-4 WMMA over [y0,y1,y2,t], accumulate onto Sc
        v2f A1; A1.x = ya.x; A1.y = half ? tt : ya.y;
        v8f h1[4];
#pragma unroll
        for (int n = 0; n < 4; ++n) {
            v8f d = wmma4(A1, w1f[n], Sc[n]);
#pragma unroll
            for (int v = 0; v < 8; ++v) d[v] = fast_tanh(d[v]);
            h1[n] = d;
        }
        // scatter C-layout -> row-major LDS stage
#pragma unroll
        for (int n = 0; n < 4; ++n)
#pragma unroll
            for (int v = 0; v < 8; ++v)
                mystage[(v + 8 * half) * LDA + n * 16 + l] = h1[n][v];
        // gather A-fragments (per-lane contiguous b64)
        v2f afr[16];
#pragma unroll
        for (int c = 0; c < 16; ++c)
            afr[c] = *(const v2f*)&mystage[l * LDA + 4 * c + 2 * half];

        // Layer 2: 16 k-chunks x 4 n-tiles of f32 WMMA, W2 B-frags streamed from LDS
        v8f h2[4];
#pragma unroll
        for (int n = 0; n < 4; ++n) {
            v8f z;
#pragma unroll
            for (int v = 0; v < 8; ++v) z[v] = b2v[n];
            h2[n] = z;
        }
#pragma unroll
        for (int c = 0; c < 16; ++c)
#pragma unroll
            for (int n = 0; n < 4; ++n) {
                float2 bf = w2s[(c * 4 + n) * 32 + lane];
                v2f b; b.x = bf.x; b.y = bf.y;
                h2[n] = wmma4(afr[c], b, h2[n]);
            }
        // tanh + scatter a2 to stage
#pragma unroll
        for (int n = 0; n < 4; ++n)
#pragma unroll
            for (int v = 0; v < 8; ++v)
                mystage[(v + 8 * half) * LDA + n * 16 + l] = fast_tanh(h2[n][v]);

        // Layer 3: a2 @ W3pad (single n-tile, 16 k-chunks)
        v8f acc3;
#pragma unroll
        for (int v = 0; v < 8; ++v) acc3[v] = b3v;
#pragma unroll
        for (int c = 0; c < 16; ++c) {
            v2f a = *(const v2f*)&mystage[l * LDA + 4 * c + 2 * half];
            acc3 = wmma4(a, w3f[c], acc3);
        }
        // k = tanh(.)*scale ; stash cols 0..3 (col3 = exact 0) into k staging
        if (l < 4) {
#pragma unroll
            for (int v = 0; v < 8; ++v)
                myk[(v + 8 * half) * 4 + l] = fast_tanh(acc3[v]) * scale;
        }
        float2 kk = *(const float2*)&myk[l * 4 + 2 * half];
        return kk;
    };

    // ---- fixed-step dopri5 ----
    for (int step = 0; step < 10; ++step) {
        const float t0 = step * hstep;
        float2 k1, k2, k3, k4, k5, k6, ya;

        k1 = feval(t0, y);
        ya = make_float2(y.x + hstep * 0.2f * k1.x,
                         y.y + hstep * 0.2f * k1.y);
        k2 = feval(t0 + 0.2f * hstep, ya);
        ya = make_float2(y.x + hstep * (0.075f * k1.x + 0.225f * k2.x),
                         y.y + hstep * (0.075f * k1.y + 0.225f * k2.y));
        k3 = feval(t0 + 0.3f * hstep, ya);
        {
            const float a1 = 44.f / 45.f, a2 = -56.f / 15.f, a3 = 32.f / 9.f;
            ya = make_float2(y.x + hstep * (a1 * k1.x + a2 * k2.x + a3 * k3.x),
                             y.y + hstep * (a1 * k1.y + a2 * k2.y + a3 * k3.y));
        }
        k4 = feval(t0 + 0.8f * hstep, ya);
        {
            const float a1 = 19372.f / 6561.f, a2 = -25360.f / 2187.f,
                        a3 = 64448.f / 6561.f, a4 = -212.f / 729.f;
            ya = make_float2(y.x + hstep * (a1 * k1.x + a2 * k2.x + a3 * k3.x + a4 * k4.x),
                             y.y + hstep * (a1 * k1.y + a2 * k2.y + a3 * k3.y + a4 * k4.y));
        }
        k5 = feval(t0 + (8.f / 9.f) * hstep, ya);
        {
            const float a1 = 9017.f / 3168.f, a2 = -355.f / 33.f, a3 = 46732.f / 5247.f,
                        a4 = 49.f / 176.f, a5 = -5103.f / 18656.f;
            ya = make_float2(y.x + hstep * (a1 * k1.x + a2 * k2.x + a3 * k3.x + a4 * k4.x + a5 * k5.x),
                             y.y + hstep * (a1 * k1.y + a2 * k2.y + a3 * k3.y + a4 * k4.y + a5 * k5.y));
        }
        k6 = feval(t0 + hstep, ya);
        {
            const float c1 = 35.f / 384.f, c3 = 500.f / 1113.f, c4 = 125.f / 192.f,
                        c5 = -2187.f / 6784.f, c6 = 11.f / 84.f;
            y = make_float2(y.x + hstep * (c1 * k1.x + c3 * k3.x + c4 * k4.x + c5 * k5.x + c6 * k6.x),
                            y.y + hstep * (c1 * k1.y + c3 * k3.y + c4 * k4.y + c5 * k5.y + c6 * k6.y));
        }
    }

    // ---- write back final PAD state ----
    float* po = out + (base + l) * 3;
    if (half) {
        po[2] = y.x;
    } else {
        po[0] = y.x;
        po[1] = y.y;
    }
}

extern "C" void kernel_launch(void* const* d_in, const int* in_sizes, int n_in,
                              void* d_out, int out_size, void* d_ws, size_t ws_size,
                              hipStream_t stream) {
    const float* pad0 = (const float*)d_in[0];
    const float* sens = (const float*)d_in[1];
    const float* W1   = (const float*)d_in[2];
    const float* b1   = (const float*)d_in[3];
    const float* W2   = (const float*)d_in[4];
    const float* b2   = (const float*)d_in[5];
    const float* W3   = (const float*)d_in[6];
    const float* b3   = (const float*)d_in[7];
    const float* sc   = (const float*)d_in[8];
    float* out = (float*)d_out;

    const int B      = in_sizes[0] / 3;
    const int ntiles = (B + TILE - 1) / TILE;
    const int blocks = (ntiles + WAVES - 1) / WAVES;

    node_rk45_kernel<<<blocks, WAVES * 32, 0, stream>>>(
        pad0, sens, W1, b1, W2, b2, W3, b3, sc, out, ntiles);
}